// RayTracing_29240137351196
// MI455X (gfx1250) — compile-verified
//
#include <hip/hip_runtime.h>
#include <math.h>

#define NSTEPS   100
#define ST_ITERS 10
#define NSECANT  8
#define SDF_R    0.5f
#define THRESH   5e-05f

struct F3 { float x, y, z; };

// sdf(o + t*d) = sqrt(qa*t^2 + qb*t + qc + 1e-12) - R, with
// qa = d.d, qb = 2*o.d, qc = o.o  (2 FMA + sqrt + sub per eval)
__device__ __forceinline__ float sdf_q(float qa, float qb, float qc, float t) {
    float s2 = fmaf(fmaf(qa, t, qb), t, qc);
    return sqrtf(s2 + 1e-12f) - SDF_R;
}

__global__ __launch_bounds__(256)
void RayTracing_29240137351196_kernel(const float* __restrict__ cam,
                                      const float* __restrict__ dir,
                                      const int*   __restrict__ objm,
                                      float* __restrict__ out_pts,
                                      float* __restrict__ out_mask,
                                      float* __restrict__ out_dist,
                                      int n) {
    int i = blockIdx.x * blockDim.x + threadIdx.x;
    if (i >= n) return;

    // Prefetch ~4K rays ahead (speculative global_prefetch_b8; OOB is dropped).
    __builtin_prefetch(cam  + 3 * (i + 4096), 0, 0);
    __builtin_prefetch(dir  + 3 * (i + 4096), 0, 0);
    __builtin_prefetch(objm +     (i + 4096), 0, 0);

    F3 o = reinterpret_cast<const F3*>(cam)[i];   // global_load_b96
    F3 d = reinterpret_cast<const F3*>(dir)[i];
    bool obj = objm[i] != 0;

    float qa = d.x*d.x + d.y*d.y + d.z*d.z;
    float qb = 2.0f * (o.x*d.x + o.y*d.y + o.z*d.z);
    float qc = o.x*o.x + o.y*o.y + o.z*o.z;

    // ---- sphere intersect with bounding sphere r=1 ----
    float c1    = qc - 1.0f;
    float under = fmaf(qb, qb, -4.0f * qa * c1);
    bool  mask  = under > 0.0f;
    float sq    = sqrtf(fmaxf(under, 0.0f));
    float nearv = fmaxf(mask ? 0.5f * (-sq - qb) : 0.0f, 0.01f);
    float farv  = fmaxf(mask ? 0.5f * ( sq - qb) : 0.0f, 0.01f);

    // ---- sphere tracing (bidirectional), 10 iters, 1 line-search step ----
    bool  unfin_s = mask, unfin_e = mask;
    float acc_s = nearv, acc_e = farv;
    float next_s = unfin_s ? sdf_q(qa, qb, qc, acc_s) : 0.0f;
    float next_e = unfin_e ? sdf_q(qa, qb, qc, acc_e) : 0.0f;
#pragma unroll 1
    for (int it = 0; it < ST_ITERS; ++it) {
        float cs = unfin_s ? next_s : 0.0f; cs = (cs <= THRESH) ? 0.0f : cs;
        float ce = unfin_e ? next_e : 0.0f; ce = (ce <= THRESH) ? 0.0f : ce;
        unfin_s = unfin_s && (cs > THRESH);
        unfin_e = unfin_e && (ce > THRESH);
        acc_s += cs;
        acc_e -= ce;
        next_s = unfin_s ? sdf_q(qa, qb, qc, acc_s) : 0.0f;
        next_e = unfin_e ? sdf_q(qa, qb, qc, acc_e) : 0.0f;
        // line search: step = (1 - 0.5)/2^0 = 0.5
        bool np_s = next_s < 0.0f;
        bool np_e = next_e < 0.0f;
        acc_s  = np_s ? fmaf(-0.5f, cs, acc_s) : acc_s;
        acc_e  = np_e ? fmaf( 0.5f, ce, acc_e) : acc_e;
        next_s = np_s ? sdf_q(qa, qb, qc, acc_s) : next_s;
        next_e = np_e ? sdf_q(qa, qb, qc, acc_e) : next_e;
        bool ok = acc_s < acc_e;
        unfin_s = unfin_s && ok;
        unfin_e = unfin_e && ok;
    }
    { // final _upd_mask on start side (end side result unused)
        float cs = unfin_s ? next_s : 0.0f; cs = (cs <= THRESH) ? 0.0f : cs;
        unfin_s = unfin_s && (cs > THRESH);
    }

    bool  net0  = acc_s < acc_e;       // net_mask before sampler override
    bool  smask = unfin_s;
    float smin  = smask ? acc_s : 0.0f;
    float smax  = smask ? acc_e : 0.0f;

    // ---- ray sampler: 100-step scan, two argmins (first-occurrence) ----
    float span = smax - smin;
    const float inv = 1.0f / (float)(NSTEPS - 1);
    float best_tmp = INFINITY; int ind  = 0;
    float best_s   = INFINITY; int oidx = 0;
#pragma unroll 1
    for (int k = 0; k < NSTEPS; ++k) {
        float z = fmaf((float)k * inv, span, smin);
        float s = sdf_q(qa, qb, qc, z);
        float w = (float)(NSTEPS - k);
        float tmp = (s > 0.0f) ? w : ((s < 0.0f) ? -w : 0.0f);
        bool bt = tmp < best_tmp;  best_tmp = bt ? tmp : best_tmp;  ind  = bt ? k : ind;
        bool bs = s   < best_s;    best_s   = bs ? s   : best_s;    oidx = bs ? k : oidx;
    }
    // recompute samples at the selected indices (sval is a pure function of z)
    float z_ind = fmaf((float)ind  * inv, span, smin);
    float s_ind = sdf_q(qa, qb, qc, z_ind);
    float z_oid = fmaf((float)oidx * inv, span, smin);
    int   ilow  = (ind > 0) ? ind - 1 : 0;
    float zl    = fmaf((float)ilow * inv, span, smin);
    float sl    = sdf_q(qa, qb, qc, zl);

    bool net_surface = s_ind < 0.0f;
    bool p_out   = smask && !(obj && net_surface);
    bool net_obj = smask && net_surface;
    bool sec     = smask && net_surface && obj;

    float dist = z_ind;
    dist = p_out ? z_oid : dist;

    // ---- secant refinement (8 iters) on (z, sdf) bracket [ilow, ind] ----
    float zh = z_ind, sh = s_ind;
    float den = sh - sl; den = (den == 0.0f) ? 1.0f : den;
    float z = zl - sl * (zh - zl) / den;
#pragma unroll 1
    for (int it = 0; it < NSECANT; ++it) {
        float s = sdf_q(qa, qb, qc, z);
        bool hi = s > 0.0f;
        zl = hi ? z : zl;  sl = hi ? s : sl;
        bool lo = s < 0.0f;
        zh = lo ? z : zh;  sh = lo ? s : sh;
        den = sh - sl; den = (den == 0.0f) ? 1.0f : den;
        z = zl - sl * (zh - zl) / den;
    }
    dist = sec ? z : dist;

    // ---- combine sampler result with tracing result ----
    // tracing curr_pts == o + acc_s*d ; sampler pts == o + dist*d
    float fdist = smask ? dist : acc_s;
    bool  fnet  = smask ? net_obj : net0;
    float px = fmaf(fdist, d.x, o.x);
    float py = fmaf(fdist, d.y, o.y);
    float pz = fmaf(fdist, d.z, o.z);

    out_pts[3 * i + 0] = px;
    out_pts[3 * i + 1] = py;
    out_pts[3 * i + 2] = pz;
    out_mask[i] = fnet ? 1.0f : 0.0f;
    out_dist[i] = fdist;
}

extern "C" void kernel_launch(void* const* d_in, const int* in_sizes, int n_in,
                              void* d_out, int out_size, void* d_ws, size_t ws_size,
                              hipStream_t stream) {
    (void)n_in; (void)d_ws; (void)ws_size; (void)out_size;
    const float* cam  = (const float*)d_in[0];   // (N,3) float32
    const float* dir  = (const float*)d_in[1];   // (N,3) float32
    const int*   objm = (const int*)d_in[2];     // (N,)  bool->int

    int n = in_sizes[0] / 3;                     // N_RAYS
    float* out_pts  = (float*)d_out;             // N*3
    float* out_mask = out_pts + 3 * (size_t)n;   // N
    float* out_dist = out_mask + (size_t)n;      // N

    dim3 block(256);
    dim3 grid((n + 255) / 256);
    RayTracing_29240137351196_kernel<<<grid, block, 0, stream>>>(
        cam, dir, objm, out_pts, out_mask, out_dist, n);
}